// VectorQuantize_22153441313349
// MI455X (gfx1250) — compile-verified
//
#include <hip/hip_runtime.h>
#include <hip/hip_bf16.h>

// Shapes (from reference): x [B=8, T=2048, H*D=512], key_embed [H=8, C=2048, D=64]
#define H 8
#define C 2048
#define D 64
#define NB 8
#define NT 2048
#define NTOK (NB * NT)      // 16384 tokens (b*t flattened)
#define HD (H * D)          // 512
#define DECAYF 0.9f

#define MT 4                // M-tiles (of 16 tokens) per wave
#define TPAD 68             // padded LDS row stride (words) -> conflict-free ds_load_b64
#define TWORDS (16 * TPAD)  // words per staged 16x64 tile

typedef float v2f __attribute__((ext_vector_type(2)));
typedef float v8f __attribute__((ext_vector_type(8)));

// ---------------------------------------------------------------------------
// Kernel 0: per-code squared norms  esq[h*C + c] = sum_d ke[h][c][d]^2.
// One wave per codebook row: coalesced 256B read + 5-step shfl_xor reduce.
// ---------------------------------------------------------------------------
__global__ __launch_bounds__(256)
void vq_code_norms(const float* __restrict__ ke, float* __restrict__ esq) {
    int row  = (blockIdx.x * 256 + threadIdx.x) >> 5;   // 0 .. H*C-1
    int lane = threadIdx.x & 31;
    v2f v = *(const v2f*)(ke + (size_t)row * D + lane * 2);
    float s = v.x * v.x + v.y * v.y;
#pragma unroll
    for (int m = 16; m >= 1; m >>= 1) s += __shfl_xor(s, m, 32);
    if (lane == 0) esq[row] = s;
}

// ---------------------------------------------------------------------------
// Kernel 1: fused distance-GEMM + argmin via V_WMMA_F32_16X16X4_F32.
// Block = 4 waves (128 thr), one head, each wave owns 4 x 16-token M-tiles.
// The 16x64 fp32 code tile is staged global->LDS once per block with
// double-buffered GLOBAL_LOAD_ASYNC_TO_LDS_B128 (ASYNCcnt), shared by all
// 4 waves; each B fragment feeds 4 WMMAs. K-loop is software-pipelined by
// one fragment pair so DS latency hides under the WMMA burst.
// Accumulators start at -0.5*|e_col|^2, so acc = dot - 0.5|e|^2 and
// argmin(dist) == argmax(acc): no per-tile norm math in the hot loop.
// ---------------------------------------------------------------------------
__global__ __launch_bounds__(128)
void vq_argmin_wmma(const float* __restrict__ x,
                    const float* __restrict__ ke,
                    const float* __restrict__ esq,
                    int* __restrict__ indOut /* [NTOK, H] int32 */) {
    __shared__ float tileBuf[2][TWORDS];            // double-buffered B tile

    const int h    = blockIdx.y;                    // head
    const int tid  = threadIdx.x;                   // 0..127
    const int wave = tid >> 5;                      // 0..3
    const int lane = tid & 31;
    const int half = lane >> 4;                     // WMMA K/N half split
    const int l16  = lane & 15;
    const int n0   = (blockIdx.x * 4 + wave) * (16 * MT);   // 64-token group

    const float* keh  = ke + (size_t)h * C * D;
    const float* esqh = esq + (size_t)h * C;

    // async-copy slice for this thread: 8 consecutive floats of the 4KB tile
    const int r  = tid >> 3;                        // tile row 0..15
    const int cc = (tid & 7) * 8;                   // col 0,8,..,56

    // ---- Preload A: a[m][kk] = flat[n0+16m+l16][4kk + 2*half .. +1] ----
    v2f a[MT][16];
#pragma unroll
    for (int m = 0; m < MT; ++m) {
        const float* p = x + (size_t)(n0 + m * 16 + l16) * HD + h * D + 2 * half;
#pragma unroll
        for (int kk = 0; kk < 16; ++kk)
            a[m][kk] = *(const v2f*)(p + 4 * kk);
    }

    float best[MT][8];
    int   bidx[MT][8];
#pragma unroll
    for (int m = 0; m < MT; ++m)
#pragma unroll
        for (int j = 0; j < 8; ++j) { best[m][j] = -3.4e38f; bidx[m][j] = 0; }

    // ---- prologue: stage tile 0 into buffer 0 ----
    {
        unsigned lds0  = (unsigned)(uintptr_t)&tileBuf[0][r * TPAD + cc];
        const float* g = keh + tid * 8;             // tile ct is contiguous 1024 floats
        asm volatile("global_load_async_to_lds_b128 %0, %1, off"
                     :: "v"(lds0), "v"(g) : "memory");
        asm volatile("global_load_async_to_lds_b128 %0, %1, off"
                     :: "v"(lds0 + 16u), "v"(g + 4) : "memory");
        asm volatile("s_wait_asynccnt 0x0" ::: "memory");
    }
    __syncthreads();

    for (int ct = 0; ct < C / 16; ++ct) {
        const int cur = ct & 1;

        // per-lane column norm (broadcast dword, L1-hot) -> C-operand init
        const float einit = -0.5f * esqh[ct * 16 + l16];

        // ---- kick off async copy of next tile into the other buffer ----
        if (ct + 1 < C / 16) {
            unsigned lds0  = (unsigned)(uintptr_t)&tileBuf[cur ^ 1][r * TPAD + cc];
            const float* g = keh + (size_t)(ct + 1) * 1024 + tid * 8;
            asm volatile("global_load_async_to_lds_b128 %0, %1, off"
                         :: "v"(lds0), "v"(g) : "memory");
            asm volatile("global_load_async_to_lds_b128 %0, %1, off"
                         :: "v"(lds0 + 16u), "v"(g + 4) : "memory");
        }

        const float* lbase = &tileBuf[cur][l16 * TPAD + 2 * half];

        // ---- WMMA burst: 4 M-tiles x 16 K-steps, C seeded with -|e|^2/2.
        //      Rolling (b0,b1) window: next pair issued before current 8 WMMAs.
        v8f acc[MT];
#pragma unroll
        for (int m = 0; m < MT; ++m)
#pragma unroll
            for (int j = 0; j < 8; ++j) acc[m][j] = einit;

        v2f b0 = *(const v2f*)(lbase + 0);
        v2f b1 = *(const v2f*)(lbase + 4);
#pragma unroll
        for (int kk = 0; kk < 16; kk += 2) {
            v2f n0v = b0, n1v = b1;
            if (kk + 2 < 16) {
                n0v = *(const v2f*)(lbase + 4 * (kk + 2));
                n1v = *(const v2f*)(lbase + 4 * (kk + 3));
            }
#pragma unroll
            for (int m = 0; m < MT; ++m)
                acc[m] = __builtin_amdgcn_wmma_f32_16x16x4_f32(
                             false, a[m][kk], false, b0, (short)0, acc[m], false, false);
#pragma unroll
            for (int m = 0; m < MT; ++m)
                acc[m] = __builtin_amdgcn_wmma_f32_16x16x4_f32(
                             false, a[m][kk + 1], false, b1, (short)0, acc[m], false, false);
            b0 = n0v;
            b1 = n1v;
        }

        const int cidx = ct * 16 + l16;
#pragma unroll
        for (int m = 0; m < MT; ++m)
#pragma unroll
            for (int j = 0; j < 8; ++j) {
                float s = acc[m][j];                // dot - 0.5|e|^2 : maximize
                bool  c = s > best[m][j];
                best[m][j] = c ? s : best[m][j];
                bidx[m][j] = c ? cidx : bidx[m][j];
            }

        // ---- close the pipeline stage: async done, all waves done reading ----
        if (ct + 1 < C / 16)
            asm volatile("s_wait_asynccnt 0x0" ::: "memory");
        __syncthreads();
    }

    // ---- Cross-lane argmax over the 16 columns held in each half-wave ----
#pragma unroll
    for (int m = 0; m < MT; ++m) {
#pragma unroll
        for (int j = 0; j < 8; ++j) {
            float s = best[m][j];
            int   i = bidx[m][j];
#pragma unroll
            for (int mask = 1; mask <= 8; mask <<= 1) {
                float os = __shfl_xor(s, mask, 32);
                int   oi = __shfl_xor(i, mask, 32);
                bool  t  = (os > s) || (os == s && oi < i);   // argmin: lowest idx on tie
                s = t ? os : s;
                i = t ? oi : i;
            }
            if (l16 == 0) {
                // C/D layout: VGPR j holds M=j (lanes 0-15) and M=j+8 (lanes 16-31)
                int tok = n0 + m * 16 + j + 8 * half;
                indOut[tok * H + h] = i;            // also the emb_ind output (int32)
            }
        }
    }
}

// ---------------------------------------------------------------------------
// Kernel 2: gather quantized vectors. One wave handles one (token, head) row:
// broadcast index load + fully coalesced 256B codebook-row copy.
// ---------------------------------------------------------------------------
__global__ __launch_bounds__(256)
void vq_gather(const float* __restrict__ ke,
               const int* __restrict__ ind,
               float* __restrict__ qout) {
    int gid  = blockIdx.x * 256 + threadIdx.x;      // 0 .. NTOK*256-1 (float2 units)
    int nn   = gid >> 8;                            // token
    int col2 = gid & 255;                           // float2 column in [0,256)
    int h    = col2 >> 5;                           // 32 float2 per head
    int dd   = (col2 & 31) * 2;
    int idx  = ind[nn * H + h];
    v2f v = *(const v2f*)(ke + ((size_t)h * C + idx) * D + dd);
    *(v2f*)(qout + (size_t)nn * HD + col2 * 2) = v;
}

// ---------------------------------------------------------------------------
// Kernel 3a: initialize new_key_embed = key_optim ? (1-DECAY)*ke : ke
// ---------------------------------------------------------------------------
__global__ __launch_bounds__(256)
void vq_update_init(const float* __restrict__ ke,
                    const int* __restrict__ key_optim,
                    float* __restrict__ out3) {
    int gid = blockIdx.x * 256 + threadIdx.x;       // 0 .. H*C*D-1
    float v = ke[gid];
    out3[gid] = (key_optim[0] != 0) ? (1.0f - DECAYF) * v : v;
}

// ---------------------------------------------------------------------------
// Kernel 3b: scatter-add DECAY * flat[h,n,:] into new_key_embed[h, ind, :]
// (global_atomic_add_f32). Wave = 32 consecutive d for one (h, token):
// broadcast index, coalesced x read, one-cacheline atomic burst.
// ---------------------------------------------------------------------------
__global__ __launch_bounds__(256)
void vq_scatter_add(const float* __restrict__ x,
                    const int* __restrict__ ind,
                    const int* __restrict__ key_optim,
                    float* __restrict__ out3) {
    if (key_optim[0] == 0) return;
    int gid = blockIdx.x * 256 + threadIdx.x;       // 0 .. H*NTOK*D-1
    int h   = gid >> 20;                            // NTOK*D = 1<<20
    int rem = gid & ((1 << 20) - 1);
    int nn  = rem >> 6;
    int dd  = rem & (D - 1);
    int idx = ind[nn * H + h];
    float v = DECAYF * x[(size_t)nn * HD + h * D + dd];
    unsafeAtomicAdd(&out3[((size_t)h * C + idx) * D + dd], v);
}

// ---------------------------------------------------------------------------
extern "C" void kernel_launch(void* const* d_in, const int* in_sizes, int n_in,
                              void* d_out, int out_size, void* d_ws, size_t ws_size,
                              hipStream_t stream) {
    (void)in_sizes; (void)n_in; (void)out_size; (void)ws_size;

    const float* x  = (const float*)d_in[0];        // [8,2048,512] f32
    const float* ke = (const float*)d_in[1];        // [8,2048,64]  f32
    const int*   ko = (const int*)d_in[2];          // key_optim scalar

    float* out   = (float*)d_out;
    float* qout  = out;                                       // [8,2048,512]
    int*   indO  = (int*)(out + (size_t)NTOK * HD);           // [8,2048,8] int32
    float* out3  = out + (size_t)NTOK * HD + (size_t)NTOK * H;// [8,2048,64]
    float* esq   = (float*)d_ws;                              // [H*C] = 64KB scratch

    vq_code_norms<<<(H * C) / 8, 256, 0, stream>>>(ke, esq);

    dim3 g1(NTOK / (4 * 16 * MT), H);               // 64 blocks x 8 heads, 4 waves each
    vq_argmin_wmma<<<g1, 128, 0, stream>>>(x, ke, esq, indO);

    vq_gather<<<(NTOK * 256) / 256, 256, 0, stream>>>(ke, indO, qout);

    vq_update_init<<<(H * C * D) / 256, 256, 0, stream>>>(ke, ko, out3);

    vq_scatter_add<<<(H * NTOK * D) / 256, 256, 0, stream>>>(x, indO, ko, out3);
}